// CoordEncoder_25958782337707
// MI455X (gfx1250) — compile-verified
//
#include <hip/hip_runtime.h>
#include <math.h>

typedef __attribute__((ext_vector_type(2))) float v2f;
typedef __attribute__((ext_vector_type(8))) float v8f;

#define NB 8
#define NL 1024
#define NE 256
#define NR 16
#define NROWS (NB * NL)
#define LOG2E 1.4426950408889634f

__device__ __forceinline__ v8f wmma_f32_k4(v2f a, v2f b, v8f c) {
  // D(16x16,f32) = A(16x4,f32) * B(4x16,f32) + C
  return __builtin_amdgcn_wmma_f32_16x16x4_f32(false, a, false, b, (short)0, c,
                                               false, false);
}

// ---------------------------------------------------------------------------
// Kernel 1: rbf_local[b, i, k] = mean_j exp(-gamma * (dist(i,j) - c_k)^2)
// One block (8 waves) per (b, 16-row tile). Gram tiles via WMMA f32 16x16x4.
// dist via ||xi||^2 + ||xj||^2 - 2 xi.xj; raw v_sqrt_f32 / v_exp_f32.
// ---------------------------------------------------------------------------
__global__ void __launch_bounds__(256)
rbf_local_kernel(const float* __restrict__ coords,
                 const float* __restrict__ centers,
                 const float* __restrict__ gamma_p,
                 float* __restrict__ rbfl) {
  const int b    = blockIdx.x >> 6;          // 8 batches
  const int i0   = (blockIdx.x & 63) << 4;   // row-tile base within L
  const int tid  = threadIdx.x;
  const int lane = tid & 31;
  const int wave = tid >> 5;                 // 0..7
  const int half = lane >> 4;                // C/D rows: half 0 -> M 0..7, half 1 -> M 8..15
  const int lid  = lane & 15;                // C/D column within tile
  const float ngl = -gamma_p[0] * LOG2E;     // exp(-g t^2) = exp2(ngl * t^2)

  __shared__ float red[NR * 16];
  red[tid] = 0.0f;
  __syncthreads();

  float cent[NR];
#pragma unroll
  for (int k = 0; k < NR; ++k) cent[k] = centers[k];

  const float* cb = coords + (size_t)b * NL * 3;

  // A fragment (16x4, M x K): lane row = lid, kbase = 2*half, k==3 is zero pad
  const int arow = i0 + lid;
  v2f afrag;
  afrag.x = cb[arow * 3 + (half ? 2 : 0)];
  afrag.y = half ? 0.0f : cb[arow * 3 + 1];

  // squared norms of the 8 rows this lane's half covers
  float rn[8];
#pragma unroll
  for (int r = 0; r < 8; ++r) {
    const int m = i0 + r + 8 * half;
    const float x = cb[m * 3 + 0];
    const float y = cb[m * 3 + 1];
    const float z = cb[m * 3 + 2];
    rn[r] = x * x + y * y + z * z;
  }

  float acc[8 * NR];
#pragma unroll
  for (int i = 0; i < 8 * NR; ++i) acc[i] = 0.0f;

  // each wave handles 8 of the 64 column tiles
  for (int jt = wave; jt < NL / 16; jt += 8) {
    const int jcol = (jt << 4) + lid;
    const float bx = cb[jcol * 3 + 0];
    const float by = cb[jcol * 3 + 1];
    const float bz = cb[jcol * 3 + 2];
    const float cn2 = bx * bx + by * by + bz * bz;

    // B fragment (4x16, K x N): lane col = lid, kbase = 2*half
    v2f bfrag;
    bfrag.x = half ? bz : bx;
    bfrag.y = half ? 0.0f : by;

    v8f dot = {};
    dot = wmma_f32_k4(afrag, bfrag, dot);  // dot[r] = x_{i0+r+8*half} . x_{jcol}

#pragma unroll
    for (int r = 0; r < 8; ++r) {
      const float sq = rn[r] + cn2 - 2.0f * dot[r];
      const float d  = __builtin_amdgcn_sqrtf(fmaxf(sq, 1e-12f));
#pragma unroll
      for (int k = 0; k < NR; ++k) {
        const float t = d - cent[k];
        acc[r * NR + k] += __builtin_amdgcn_exp2f(ngl * t * t);
      }
    }
  }

  // reduce over the 16 columns held by each half-wave
#pragma unroll
  for (int i = 0; i < 8 * NR; ++i) {
    float v = acc[i];
    v += __shfl_xor(v, 1, 16);
    v += __shfl_xor(v, 2, 16);
    v += __shfl_xor(v, 4, 16);
    v += __shfl_xor(v, 8, 16);
    acc[i] = v;
  }
  if (lid == 0) {
#pragma unroll
    for (int r = 0; r < 8; ++r)
#pragma unroll
      for (int k = 0; k < NR; ++k)
        atomicAdd(&red[(r + 8 * half) * NR + k], acc[r * NR + k]);
  }
  __syncthreads();

  // 256 threads write the 16x16 tile of rbf_local (mean over L neighbors)
  {
    const int m = tid >> 4;
    const int k = tid & 15;
    rbfl[((size_t)b * NL + i0 + m) * NR + k] = red[tid] * (1.0f / (float)NL);
  }
}

// ---------------------------------------------------------------------------
// Kernel 2: Weff[k,n] = sum_e W1[k,e] * W2[256+e, n]
//           beff[n]  = b2[n] + sum_e b1[e] * W2[256+e, n]
// ---------------------------------------------------------------------------
__global__ void __launch_bounds__(256)
weff_kernel(const float* __restrict__ w1, const float* __restrict__ b1,
            const float* __restrict__ w2, const float* __restrict__ b2,
            float* __restrict__ weff, float* __restrict__ beff) {
  const int idx = blockIdx.x * blockDim.x + threadIdx.x;
  if (idx < NR * NE) {
    const int k = idx >> 8;
    const int n = idx & (NE - 1);
    float s = 0.0f;
    for (int e = 0; e < NE; ++e) s += w1[k * NE + e] * w2[(NE + e) * NE + n];
    weff[idx] = s;
  } else if (idx < NR * NE + NE) {
    const int n = idx - NR * NE;
    float s = b2[n];
    for (int e = 0; e < NE; ++e) s += b1[e] * w2[(NE + e) * NE + n];
    beff[n] = s;
  }
}

// ---------------------------------------------------------------------------
// Kernel 3: out[m,n] = sum_e atom_emb[Z[m],e]*W2[e,n]
//                    + sum_k rbfl[m,k]*Weff[k,n] + beff[n]
// One wave per 16x16 output tile; K accumulated with WMMA f32 16x16x4.
// ---------------------------------------------------------------------------
__global__ void __launch_bounds__(256)
fused_out_kernel(const int* __restrict__ Z,
                 const float* __restrict__ atom_emb,
                 const float* __restrict__ w2,
                 const float* __restrict__ rbfl,
                 const float* __restrict__ weff,
                 const float* __restrict__ beff,
                 float* __restrict__ out) {
  const int lane = threadIdx.x & 31;
  const int wave = threadIdx.x >> 5;
  const int half = lane >> 4;
  const int lid  = lane & 15;
  const int gw   = blockIdx.x * 8 + wave;  // global wave id: 0..8191
  const int tm   = gw >> 4;                // 512 row tiles
  const int tn   = gw & 15;                // 16 col tiles
  const int row0 = tm << 4;
  const int jcol = (tn << 4) + lid;

  const int zrow = Z[row0 + lid];
  const float* aemb = atom_emb + (size_t)zrow * NE;
  const float* arbf = rbfl + (size_t)(row0 + lid) * NR;

  const float bval = beff[jcol];
  v8f c;
#pragma unroll
  for (int r = 0; r < 8; ++r) c[r] = bval;

  // h_atom @ W2[:256]  (K = 256 in 64 steps of 4)
#pragma unroll 4
  for (int s = 0; s < NE / 4; ++s) {
    const int kb = 4 * s + 2 * half;
    v2f a;
    a.x = aemb[kb];
    a.y = aemb[kb + 1];
    v2f bb;
    bb.x = w2[kb * NE + jcol];
    bb.y = w2[(kb + 1) * NE + jcol];
    c = wmma_f32_k4(a, bb, c);
  }
  // rbf_local @ Weff  (K = 16 in 4 steps of 4)
#pragma unroll
  for (int s = 0; s < NR / 4; ++s) {
    const int kb = 4 * s + 2 * half;
    v2f a;
    a.x = arbf[kb];
    a.y = arbf[kb + 1];
    v2f bb;
    bb.x = weff[kb * NE + jcol];
    bb.y = weff[(kb + 1) * NE + jcol];
    c = wmma_f32_k4(a, bb, c);
  }

#pragma unroll
  for (int r = 0; r < 8; ++r)
    out[(size_t)(row0 + r + 8 * half) * NE + jcol] = c[r];
}

// ---------------------------------------------------------------------------
extern "C" void kernel_launch(void* const* d_in, const int* in_sizes, int n_in,
                              void* d_out, int out_size, void* d_ws,
                              size_t ws_size, hipStream_t stream) {
  const float* coords   = (const float*)d_in[0];  // [8,1024,3]
  const int*   Z        = (const int*)d_in[1];    // [8,1024] int32
  const float* atom_emb = (const float*)d_in[2];  // [118,256]
  const float* centers  = (const float*)d_in[3];  // [16]
  const float* gamma    = (const float*)d_in[4];  // [1]
  const float* w1       = (const float*)d_in[5];  // [16,256]
  const float* b1       = (const float*)d_in[6];  // [256]
  const float* w2       = (const float*)d_in[7];  // [512,256]
  const float* b2       = (const float*)d_in[8];  // [256]
  float* outp = (float*)d_out;                    // [8,1024,256]

  // workspace carve-out (~540 KB)
  float* rbfl = (float*)d_ws;            // 8192 * 16
  float* weff = rbfl + NROWS * NR;       // 16 * 256
  float* beff = weff + NR * NE;          // 256

  weff_kernel<<<(NR * NE + NE + 255) / 256, 256, 0, stream>>>(w1, b1, w2, b2,
                                                              weff, beff);
  rbf_local_kernel<<<NB * (NL / 16), 256, 0, stream>>>(coords, centers, gamma,
                                                       rbfl);
  fused_out_kernel<<<(NROWS / 16) * (NE / 16) / 8, 256, 0, stream>>>(
      Z, atom_emb, w2, rbfl, weff, beff, outp);
}